// GeometryAdaptiveSampling_29841432773195
// MI455X (gfx1250) — compile-verified
//
#include <hip/hip_runtime.h>
#include <hip/hip_bf16.h>

// ---------------------------------------------------------------------------
// GeometryAdaptiveSampling on MI455X (gfx1250, wave32)
//
// KNN dominates (B*N^2 = 2^31 dot products, K=3). xyz = 786KB -> L2 resident,
// compute/selection bound: V_WMMA_F32_16X16X4_F32 computes a 16x16 tile of
// dot products per instruction (K padded 3->4 with zeros). Support points are
// staged into LDS in 1024-point chunks by the Tensor Data Mover
// (tensor_load_to_lds, TENSORcnt), double-buffered to overlap DMA with the
// WMMA + register top-16 selection. MLPs are memory-trivial; BN stats via LDS
// float atomics. Top-k = deterministic radix select + ordered compaction.
// ---------------------------------------------------------------------------

#define BB 2
#define NN 32768
#define CC 64
#define KNN_K 16
#define LK 4096          // learned_k = 8192 * (1 - 0.5)
#define K2 8192          // combined
#define BN_TOT (BB * NN)
#define EPSF 1e-5f
#define CH 1024          // support points staged per TDM chunk

typedef __attribute__((ext_vector_type(2))) float v2f;
typedef __attribute__((ext_vector_type(8))) float v8f;
typedef __attribute__((ext_vector_type(4))) unsigned int u32x4;
typedef __attribute__((ext_vector_type(4))) int i32x4;
typedef __attribute__((ext_vector_type(8))) int i32x8;

__device__ __forceinline__ float lrelu(float x) { return x > 0.f ? x : 0.2f * x; }

// map float -> order-preserving unsigned
__device__ __forceinline__ unsigned f2ord(float f) {
  unsigned b = __float_as_uint(f);
  return (b & 0x80000000u) ? ~b : (b | 0x80000000u);
}

// Generic LDS pointer -> LDS byte offset (ISA: LDS aperture addr[31:0]).
__device__ __forceinline__ unsigned lds_off(const void* p) {
  return (unsigned)(uintptr_t)p;
}

// ---------------------------------------------------------------------------
// TDM: 1-D contiguous copy of nelem4 dwords global->LDS (D# per ISA 08).
// group0: count=1 | lds_addr | global_addr[56:0] | type=2
// group1: data_size=4B, tensor_dim0=tile_dim0=nelem4 (no OOB), dims1/2 unused
// groups 2/3 zero (<=2D tensor). All operands block-uniform -> SGPRs.
// ---------------------------------------------------------------------------
__device__ __forceinline__ void tdm_load_1d(unsigned lds_byte_addr,
                                            const void* gptr, unsigned nelem4) {
  unsigned long long ga = (unsigned long long)(uintptr_t)gptr;
  u32x4 g0;
  g0[0] = 1u;                                               // count=1 (valid)
  g0[1] = lds_byte_addr;                                    // LDS dest
  g0[2] = (unsigned)(ga & 0xFFFFFFFFu);                     // global_addr lo
  g0[3] = (unsigned)((ga >> 32) & 0x01FFFFFFu) | (2u << 30);// addr hi | type=2
  i32x8 g1;
  g1[0] = (int)(2u << 16);                                  // data_size=2 (4B)
  g1[1] = (int)((nelem4 & 0xFFFFu) << 16);                  // tensor_dim0 lo16
  g1[2] = (int)(((nelem4 >> 16) & 0xFFFFu) | (1u << 16));   // dim0 hi | dim1=1
  g1[3] = (int)((nelem4 & 0xFFFFu) << 16);                  // tile_dim0
  g1[4] = 1;                                                // tile_dim1=1
  g1[5] = (int)nelem4;                                      // dim0_stride lo
  g1[6] = 0;
  g1[7] = 0;
  i32x4 g2 = {0, 0, 0, 0};
  i32x4 g3 = {0, 0, 0, 0};
#if defined(__clang_major__) && __clang_major__ >= 23
  i32x8 g4 = {0, 0, 0, 0, 0, 0, 0, 0};
  __builtin_amdgcn_tensor_load_to_lds(g0, g1, g2, g3, g4, 0);
#else
  __builtin_amdgcn_tensor_load_to_lds(g0, g1, g2, g3, 0);
#endif
}

// ---------------------------------------------------------------------------
// K-zero: clear selection flags + BN stat accumulators each call
// ---------------------------------------------------------------------------
__global__ __launch_bounds__(256) void zero_kernel(int* __restrict__ flag,
                                                   float* __restrict__ stats,
                                                   int nflag, int nstats) {
  int i = blockIdx.x * 256 + threadIdx.x;
  if (i < nflag) flag[i] = 0;
  if (i < nstats) stats[i] = 0.f;
}

// ---------------------------------------------------------------------------
// K1: fused WMMA KNN (16-NN) + 3x3 local covariance.
// Block = 256 threads = 8 waves; each wave owns 32 consecutive queries.
// TDM double-buffers 1024-point xyz chunks into LDS (wave0 issues, tensorcnt
// synchronized, block barrier publishes). Per 16-point support tile: two
// V_WMMA_F32_16X16X4_F32 -> 32x16 dot tile; dist_cmp = ||p||^2 - 2 q.p into a
// padded LDS tile; each lane (one query) scans 16 candidates into a register
// top-16 (argmax-replace, fully unrolled -> no scratch).
// A layout (f32 16x4):  lanes 0-15 (K0,K1)=(x,y), lanes 16-31 (K2,K3)=(z,0)
// B layout (f32 4x16):  v0: lanes0-15 K0=x, lanes16-31 K1=y ; v1: K2=z, K3=0
// D layout: elem v of lane l = row (v + 8*(l>=16)), col l%16
// ---------------------------------------------------------------------------
__global__ __launch_bounds__(256) void knn_cov_kernel(const float* __restrict__ xyz,
                                                      float* __restrict__ cov) {
  __shared__ float sxyz[2][CH * 3];   // double-buffered support chunk (24KB)
  __shared__ float dtile[8][32][17];  // per-wave 32x16 dist tile, padded (17KB)

  const int blocks_per_batch = NN / 256;
  int b = blockIdx.x / blocks_per_batch;
  int qblk = (blockIdx.x % blocks_per_batch) * 256;
  int wave = threadIdx.x >> 5;
  int lane = threadIdx.x & 31;
  int qbase = qblk + wave * 32;
  int m = lane & 15;
  bool hi = lane >= 16;

  const float* X = xyz + (size_t)b * NN * 3;

  // A operands for the two 16-query tiles (loop invariant)
  v2f a0, a1;
  {
    const float* p0 = X + (size_t)(qbase + m) * 3;
    const float* p1 = X + (size_t)(qbase + 16 + m) * 3;
    a0.x = hi ? p0[2] : p0[0];
    a0.y = hi ? 0.f : p0[1];
    a1.x = hi ? p1[2] : p1[0];
    a1.y = hi ? 0.f : p1[1];
  }

  float kd[KNN_K];
  int ki[KNN_K];
#pragma unroll
  for (int t = 0; t < KNN_K; ++t) { kd[t] = 3.0e38f; ki[t] = 0; }
  float kworst = 3.0e38f;
  int kwslot = 0;

  int myq = qbase + lane;
  const int nchunks = NN / CH;

  // prologue: stage chunk 0
  if (threadIdx.x < 32) {
    tdm_load_1d(lds_off(&sxyz[0][0]), X, CH * 3);
    __builtin_amdgcn_s_wait_tensorcnt(0);
  }
  __syncthreads();

  for (int c = 0; c < nchunks; ++c) {
    int buf = c & 1;
    // kick DMA for next chunk; overlaps with this chunk's compute
    if (threadIdx.x < 32 && (c + 1) < nchunks) {
      tdm_load_1d(lds_off(&sxyz[buf ^ 1][0]), X + (size_t)(c + 1) * CH * 3,
                  CH * 3);
    }
    const float* S = &sxyz[buf][0];

    for (int jj = 0; jj < CH; jj += 16) {
      // B operand + ||p||^2 from LDS (unconditional, conflict-free reads)
      const float* p = S + (size_t)(jj + m) * 3;
      float px = p[0], py = p[1], pz = p[2];
      v2f bm;
      bm.x = hi ? py : px;  // K0=x (lo) / K1=y (hi)
      bm.y = hi ? 0.f : pz; // K2=z (lo) / K3=0 (hi)
      float sq = px * px + py * py + pz * pz;

      v8f c0 = {};
      v8f c1 = {};
      c0 = __builtin_amdgcn_wmma_f32_16x16x4_f32(false, a0, false, bm, (short)0,
                                                 c0, false, false);
      c1 = __builtin_amdgcn_wmma_f32_16x16x4_f32(false, a1, false, bm, (short)0,
                                                 c1, false, false);

      int rbase = hi ? 8 : 0;
#pragma unroll
      for (int v = 0; v < 8; ++v) {
        dtile[wave][rbase + v][m] = sq - 2.0f * c0[v];
        dtile[wave][16 + rbase + v][m] = sq - 2.0f * c1[v];
      }
      __syncthreads();

      int jglob = c * CH + jj;
#pragma unroll
      for (int n = 0; n < 16; ++n) {
        float d = dtile[wave][lane][n];
        if (d < kworst) {
          int idx = jglob + n;
#pragma unroll
          for (int t = 0; t < KNN_K; ++t)
            if (t == kwslot) { kd[t] = d; ki[t] = idx; }
          kworst = kd[0];
          kwslot = 0;
#pragma unroll
          for (int t = 1; t < KNN_K; ++t)
            if (kd[t] > kworst) { kworst = kd[t]; kwslot = t; }
        }
      }
      __syncthreads();
    }

    // ensure next chunk's DMA landed before buffers swap
    if (threadIdx.x < 32 && (c + 1) < nchunks) {
      __builtin_amdgcn_s_wait_tensorcnt(0);
    }
    __syncthreads();
  }

  // fused local covariance over the 16 neighbours (order-invariant sum)
  const float* q = X + (size_t)myq * 3;
  float qx = q[0], qy = q[1], qz = q[2];
  float cxx = 0, cxy = 0, cxz = 0, cyy = 0, cyz = 0, czz = 0;
#pragma unroll
  for (int t = 0; t < KNN_K; ++t) {
    const float* pp = X + (size_t)ki[t] * 3;
    float dx = pp[0] - qx, dy = pp[1] - qy, dz = pp[2] - qz;
    cxx += dx * dx; cxy += dx * dy; cxz += dx * dz;
    cyy += dy * dy; cyz += dy * dz; czz += dz * dz;
  }
  const float s = 1.0f / (float)KNN_K;
  float* o = cov + ((size_t)b * NN + myq) * 9;
  o[0] = cxx * s; o[1] = cxy * s; o[2] = cxz * s;
  o[3] = cxy * s; o[4] = cyy * s; o[5] = cyz * s;
  o[6] = cxz * s; o[7] = cyz * s; o[8] = czz * s;
}

// ---------------------------------------------------------------------------
// Generic MLP layer: optional (BN+lrelu) on input, then x @ W + b.
// Optionally accumulates column sum / sumsq of the *raw* output (for the next
// layer's BatchNorm) via LDS float atomics -> one global atomic per column.
// ---------------------------------------------------------------------------
template <int IN, int OUT>
__global__ __launch_bounds__(256) void layer_kernel(
    const float* __restrict__ in, const float* __restrict__ stats_in,
    const float* __restrict__ gamma, const float* __restrict__ beta,
    const float* __restrict__ w, const float* __restrict__ bias,
    float* __restrict__ out, float* __restrict__ stats_out, int rows) {
  __shared__ float sw[IN * OUT];
  __shared__ float sscale[IN], sshift[IN];
  __shared__ float csum[OUT], csumsq[OUT];

  for (int t = threadIdx.x; t < IN * OUT; t += 256) sw[t] = w[t];
  if (stats_in && threadIdx.x < IN) {
    float inv = 1.0f / (float)rows;
    float mean = stats_in[threadIdx.x] * inv;
    float var = stats_in[IN + threadIdx.x] * inv - mean * mean;
    float sc = gamma[threadIdx.x] * rsqrtf(var + EPSF);
    sscale[threadIdx.x] = sc;
    sshift[threadIdx.x] = beta[threadIdx.x] - mean * sc;
  }
  if (stats_out && threadIdx.x < OUT) { csum[threadIdx.x] = 0.f; csumsq[threadIdx.x] = 0.f; }
  __syncthreads();

  int row = blockIdx.x * 256 + threadIdx.x;
  if (row < rows) {
    float x[IN];
#pragma unroll
    for (int i = 0; i < IN; ++i) {
      float v = in[(size_t)row * IN + i];
      if (stats_in) v = lrelu(v * sscale[i] + sshift[i]);
      x[i] = v;
    }
    for (int o = 0; o < OUT; ++o) {
      float acc = bias[o];
#pragma unroll
      for (int i = 0; i < IN; ++i) acc += x[i] * sw[i * OUT + o];
      out[(size_t)row * OUT + o] = acc;
      if (stats_out) {
        atomicAdd(&csum[o], acc);
        atomicAdd(&csumsq[o], acc * acc);
      }
    }
  }
  __syncthreads();
  if (stats_out && threadIdx.x < OUT) {
    atomicAdd(&stats_out[threadIdx.x], csum[threadIdx.x]);
    atomicAdd(&stats_out[OUT + threadIdx.x], csumsq[threadIdx.x]);
  }
}

// Boundary layer 1: input is concat(xyz, features) built on the fly (IN=67).
__global__ __launch_bounds__(256) void bnd1_kernel(
    const float* __restrict__ xyz, const float* __restrict__ feat,
    const float* __restrict__ w, const float* __restrict__ bias,
    float* __restrict__ out, float* __restrict__ stats_out, int rows) {
  const int IN = CC + 3, OUT = 64;
  __shared__ float sw[(CC + 3) * 64];
  __shared__ float csum[64], csumsq[64];
  for (int t = threadIdx.x; t < IN * OUT; t += 256) sw[t] = w[t];
  if (threadIdx.x < OUT) { csum[threadIdx.x] = 0.f; csumsq[threadIdx.x] = 0.f; }
  __syncthreads();

  int row = blockIdx.x * 256 + threadIdx.x;
  if (row < rows) {
    float x[CC + 3];
#pragma unroll
    for (int i = 0; i < 3; ++i) x[i] = xyz[(size_t)row * 3 + i];
#pragma unroll
    for (int i = 0; i < CC; ++i) x[3 + i] = feat[(size_t)row * CC + i];
    for (int o = 0; o < OUT; ++o) {
      float acc = bias[o];
#pragma unroll
      for (int i = 0; i < IN; ++i) acc += x[i] * sw[i * OUT + o];
      out[(size_t)row * OUT + o] = acc;
      atomicAdd(&csum[o], acc);
      atomicAdd(&csumsq[o], acc * acc);
    }
  }
  __syncthreads();
  if (threadIdx.x < OUT) {
    atomicAdd(&stats_out[threadIdx.x], csum[threadIdx.x]);
    atomicAdd(&stats_out[OUT + threadIdx.x], csumsq[threadIdx.x]);
  }
}

// ---------------------------------------------------------------------------
// Score fusion: imp = lrelu([curv,bnd]@fw1+fb1)@fw2+fb2 ; gumbel score bits.
// ---------------------------------------------------------------------------
__global__ __launch_bounds__(256) void fuse_kernel(
    const float* __restrict__ curv, const float* __restrict__ bnd,
    const float* __restrict__ fw1, const float* __restrict__ fb1,
    const float* __restrict__ fw2, const float* __restrict__ fb2,
    const float* __restrict__ gum, const float* __restrict__ tau,
    float* __restrict__ imp, unsigned* __restrict__ score_u) {
  int i = blockIdx.x * 256 + threadIdx.x;
  if (i >= BN_TOT) return;
  float a = curv[i], c = bnd[i];
  float acc = fb2[0];
#pragma unroll
  for (int j = 0; j < 16; ++j) {
    float h = lrelu(a * fw1[j] + c * fw1[16 + j] + fb1[j]);
    acc += h * fw2[j];
  }
  imp[i] = acc;
  // softmax is monotone: top-k of y_soft == top-k of (imp+g)/tau
  score_u[i] = f2ord((acc + gum[i]) / tau[0]);
}

// ---------------------------------------------------------------------------
// Softmax over N per batch (for the softmax(imp) output only).
// ---------------------------------------------------------------------------
__global__ __launch_bounds__(1024) void softmax_stats_kernel(
    const float* __restrict__ imp, float* __restrict__ mOut, float* __restrict__ sOut) {
  __shared__ float red[1024];
  int b = blockIdx.x;
  const float* x = imp + (size_t)b * NN;
  float m = -3.4e38f;
  for (int i = threadIdx.x; i < NN; i += 1024) m = fmaxf(m, x[i]);
  red[threadIdx.x] = m;
  __syncthreads();
  for (int off = 512; off > 0; off >>= 1) {
    if (threadIdx.x < off) red[threadIdx.x] = fmaxf(red[threadIdx.x], red[threadIdx.x + off]);
    __syncthreads();
  }
  m = red[0];
  __syncthreads();
  float s = 0.f;
  for (int i = threadIdx.x; i < NN; i += 1024) s += __expf(x[i] - m);
  red[threadIdx.x] = s;
  __syncthreads();
  for (int off = 512; off > 0; off >>= 1) {
    if (threadIdx.x < off) red[threadIdx.x] += red[threadIdx.x + off];
    __syncthreads();
  }
  if (threadIdx.x == 0) { mOut[b] = m; sOut[b] = red[0]; }
}

__global__ __launch_bounds__(256) void softmax_write_kernel(
    const float* __restrict__ imp, const float* __restrict__ m,
    const float* __restrict__ s, float* __restrict__ out) {
  int i = blockIdx.x * 256 + threadIdx.x;
  if (i >= BN_TOT) return;
  int b = i / NN;
  out[i] = __expf(imp[i] - m[b]) / s[b];
}

// ---------------------------------------------------------------------------
// Deterministic radix select (MSB->LSB) of the k-th largest orderable uint.
// One block per batch. Outputs threshold T and #ties to take (lowest-index).
// ---------------------------------------------------------------------------
__global__ __launch_bounds__(1024) void radix_select_kernel(
    const unsigned* __restrict__ u, int k, unsigned* __restrict__ Tout,
    int* __restrict__ tieOut) {
  __shared__ int cnt;
  int b = blockIdx.x;
  const unsigned* ub = u + (size_t)b * NN;
  unsigned prefix = 0;
  int need = k;
  for (int bit = 31; bit >= 0; --bit) {
    if (threadIdx.x == 0) cnt = 0;
    __syncthreads();
    unsigned mask = 0xFFFFFFFFu << bit;   // bits [31:bit]
    unsigned want = prefix | (1u << bit);
    int c = 0;
    for (int i = threadIdx.x; i < NN; i += 1024)
      if ((ub[i] & mask) == want) ++c;
    atomicAdd(&cnt, c);
    __syncthreads();
    int tot = cnt;
    if (tot >= need) prefix |= (1u << bit);
    else need -= tot;
    __syncthreads();
  }
  // count strictly greater than T
  if (threadIdx.x == 0) cnt = 0;
  __syncthreads();
  int g = 0;
  for (int i = threadIdx.x; i < NN; i += 1024)
    if (ub[i] > prefix) ++g;
  atomicAdd(&cnt, g);
  __syncthreads();
  if (threadIdx.x == 0) { Tout[b] = prefix; tieOut[b] = k - cnt; }
}

// ---------------------------------------------------------------------------
// Ordered compaction of {u > T} U first tie_take ties (by index) -> ascending
// index list + selection flags. One block per batch; packed block scans.
// ---------------------------------------------------------------------------
__global__ __launch_bounds__(1024) void compact_threshold_kernel(
    const unsigned* __restrict__ u, const unsigned* __restrict__ Tarr,
    const int* __restrict__ tieArr, int* __restrict__ outlist,
    float* __restrict__ outf, int kcap, int* __restrict__ flag) {
  __shared__ int tmp[1024];
  __shared__ int sbase[2];  // [0]=output base, [1]=ties seen
  int b = blockIdx.x;
  const unsigned* ub = u + (size_t)b * NN;
  unsigned T = Tarr[b];
  int tie_take = tieArr[b];
  if (threadIdx.x == 0) { sbase[0] = 0; sbase[1] = 0; }
  __syncthreads();

  for (int start = 0; start < NN; start += 1024) {
    int i = start + threadIdx.x;
    unsigned x = ub[i];
    int p1 = (x > T) ? 1 : 0;
    int p2 = (x == T) ? 1 : 0;
    int packed = (p1 << 16) | p2;
    tmp[threadIdx.x] = packed;
    __syncthreads();
    for (int off = 1; off < 1024; off <<= 1) {
      int t = threadIdx.x >= off ? tmp[threadIdx.x - off] : 0;
      __syncthreads();
      tmp[threadIdx.x] += t;
      __syncthreads();
    }
    int incl = tmp[threadIdx.x];
    int tot = tmp[1023];
    int excl = incl - packed;
    int gt_before = excl >> 16;
    int tie_before = excl & 0xFFFF;
    int base = sbase[0], tiebase = sbase[1];
    int rem = tie_take - tiebase;
    if (rem < 0) rem = 0;
    int take = p1 | (p2 & (tie_before < rem ? 1 : 0));
    if (take) {
      int pos = base + gt_before + (tie_before < rem ? tie_before : rem);
      if (pos < kcap) {
        if (outlist) outlist[(size_t)b * kcap + pos] = i;
        if (outf) outf[(size_t)b * kcap + pos] = (float)i;
        flag[(size_t)b * NN + i] = 1;
      }
    }
    __syncthreads();
    if (threadIdx.x == 0) {
      int gt_tot = tot >> 16, tie_tot = tot & 0xFFFF;
      int taken = tie_tot < rem ? tie_tot : rem;
      sbase[0] = base + gt_tot + taken;
      sbase[1] = tiebase + tie_tot;
    }
    __syncthreads();
  }
}

// Ordered compaction of flag!=0 -> combined_idx (already sorted ascending).
__global__ __launch_bounds__(1024) void compact_flag_kernel(
    const int* __restrict__ flag, int* __restrict__ outlist,
    float* __restrict__ outf, int kcap) {
  __shared__ int tmp[1024];
  __shared__ int sbase;
  int b = blockIdx.x;
  const int* fb = flag + (size_t)b * NN;
  if (threadIdx.x == 0) sbase = 0;
  __syncthreads();
  for (int start = 0; start < NN; start += 1024) {
    int i = start + threadIdx.x;
    int p = fb[i] ? 1 : 0;
    tmp[threadIdx.x] = p;
    __syncthreads();
    for (int off = 1; off < 1024; off <<= 1) {
      int t = threadIdx.x >= off ? tmp[threadIdx.x - off] : 0;
      __syncthreads();
      tmp[threadIdx.x] += t;
      __syncthreads();
    }
    int incl = tmp[threadIdx.x];
    int tot = tmp[1023];
    int base = sbase;
    if (p) {
      int pos = base + (incl - p);
      if (pos < kcap) {
        outlist[(size_t)b * kcap + pos] = i;
        outf[(size_t)b * kcap + pos] = (float)i;
      }
    }
    __syncthreads();
    if (threadIdx.x == 0) sbase = base + tot;
    __syncthreads();
  }
}

// rand scores with learned positions forced to -1, mapped to orderable bits
__global__ __launch_bounds__(256) void rand_mask_kernel(
    const float* __restrict__ rnd, const int* __restrict__ flag,
    unsigned* __restrict__ u2) {
  int i = blockIdx.x * 256 + threadIdx.x;
  if (i >= BN_TOT) return;
  u2[i] = f2ord(flag[i] ? -1.0f : rnd[i]);
}

// gather sampled features / xyz
__global__ __launch_bounds__(256) void gather_kernel(
    const float* __restrict__ xyz, const float* __restrict__ feat,
    const int* __restrict__ comb, float* __restrict__ of, float* __restrict__ ox) {
  int t = blockIdx.x * 256 + threadIdx.x;
  if (t >= BB * K2) return;
  int b = t / K2;
  int idx = comb[t];
  size_t src = (size_t)b * NN + idx;
  for (int c = 0; c < CC; ++c) of[(size_t)t * CC + c] = feat[src * CC + c];
  ox[t * 3 + 0] = xyz[src * 3 + 0];
  ox[t * 3 + 1] = xyz[src * 3 + 1];
  ox[t * 3 + 2] = xyz[src * 3 + 2];
}

// ---------------------------------------------------------------------------
extern "C" void kernel_launch(void* const* d_in, const int* in_sizes, int n_in,
                              void* d_out, int out_size, void* d_ws, size_t ws_size,
                              hipStream_t stream) {
  const float* xyz = (const float*)d_in[0];
  const float* feat = (const float*)d_in[1];
  const float* gum = (const float*)d_in[2];
  const float* rnd = (const float*)d_in[3];
  const float* tau = (const float*)d_in[4];
  const float* cw1 = (const float*)d_in[5];
  const float* cb1 = (const float*)d_in[6];
  const float* cg1 = (const float*)d_in[7];
  const float* cbe1 = (const float*)d_in[8];
  const float* cw2 = (const float*)d_in[9];
  const float* cb2 = (const float*)d_in[10];
  const float* bw1 = (const float*)d_in[11];
  const float* bb1 = (const float*)d_in[12];
  const float* bg1 = (const float*)d_in[13];
  const float* bbe1 = (const float*)d_in[14];
  const float* bw2 = (const float*)d_in[15];
  const float* bb2 = (const float*)d_in[16];
  const float* bg2 = (const float*)d_in[17];
  const float* bbe2 = (const float*)d_in[18];
  const float* bw3 = (const float*)d_in[19];
  const float* bb3 = (const float*)d_in[20];
  const float* fw1 = (const float*)d_in[21];
  const float* fb1 = (const float*)d_in[22];
  const float* fw2 = (const float*)d_in[23];
  const float* fb2 = (const float*)d_in[24];

  // workspace layout
  float* w = (float*)d_ws;
  float* cov = w;                                   // BN*9
  float* h1c = cov + (size_t)BN_TOT * 9;            // BN*32
  float* h1b = h1c + (size_t)BN_TOT * 32;           // BN*64
  float* h2b = h1b + (size_t)BN_TOT * 64;           // BN*32
  float* curv = h2b + (size_t)BN_TOT * 32;          // BN
  float* bnd = curv + BN_TOT;                       // BN
  float* imp = bnd + BN_TOT;                        // BN
  unsigned* score_u = (unsigned*)(imp + BN_TOT);    // BN
  unsigned* rand_u = score_u + BN_TOT;              // BN
  int* flag = (int*)(rand_u + BN_TOT);              // BN
  int* learned_list = flag + BN_TOT;                // B*LK
  int* comb_list = learned_list + BB * LK;          // B*K2
  unsigned* Tl = (unsigned*)(comb_list + BB * K2);  // B
  int* ttl = (int*)(Tl + BB);                       // B
  unsigned* Tr = (unsigned*)(ttl + BB);             // B
  int* ttr = (int*)(Tr + BB);                       // B
  float* smax = (float*)(ttr + BB);                 // B
  float* ssum = smax + BB;                          // B
  float* stats_c1 = ssum + BB;                      // 64
  float* stats_b1 = stats_c1 + 64;                  // 128
  float* stats_b2 = stats_b1 + 128;                 // 64   (stats total 256)

  // output layout (return-order concat)
  float* o_feat = (float*)d_out;                    // B*K2*CC
  float* o_xyz = o_feat + (size_t)BB * K2 * CC;     // B*K2*3
  float* o_comb = o_xyz + (size_t)BB * K2 * 3;      // B*K2
  float* o_soft = o_comb + (size_t)BB * K2;         // B*N
  float* o_lidx = o_soft + (size_t)BB * NN;         // B*LK

  const int RB = BN_TOT / 256;  // 256 row-blocks

  zero_kernel<<<RB, 256, 0, stream>>>(flag, stats_c1, BN_TOT, 256);

  knn_cov_kernel<<<BB * (NN / 256), 256, 0, stream>>>(xyz, cov);

  // curvature MLP: 9 -> 32 (stats) -> BN+lrelu -> 1
  layer_kernel<9, 32><<<RB, 256, 0, stream>>>(cov, nullptr, nullptr, nullptr,
                                              cw1, cb1, h1c, stats_c1, BN_TOT);
  // boundary MLP: 67 -> 64 (stats) -> BN+lrelu -> 32 (stats) -> BN+lrelu -> 1
  bnd1_kernel<<<RB, 256, 0, stream>>>(xyz, feat, bw1, bb1, h1b, stats_b1, BN_TOT);
  layer_kernel<32, 1><<<RB, 256, 0, stream>>>(h1c, stats_c1, cg1, cbe1,
                                              cw2, cb2, curv, nullptr, BN_TOT);
  layer_kernel<64, 32><<<RB, 256, 0, stream>>>(h1b, stats_b1, bg1, bbe1,
                                               bw2, bb2, h2b, stats_b2, BN_TOT);
  layer_kernel<32, 1><<<RB, 256, 0, stream>>>(h2b, stats_b2, bg2, bbe2,
                                              bw3, bb3, bnd, nullptr, BN_TOT);

  fuse_kernel<<<RB, 256, 0, stream>>>(curv, bnd, fw1, fb1, fw2, fb2,
                                      gum, tau, imp, score_u);

  softmax_stats_kernel<<<BB, 1024, 0, stream>>>(imp, smax, ssum);
  softmax_write_kernel<<<RB, 256, 0, stream>>>(imp, smax, ssum, o_soft);

  // learned top-4096 (ascending) + flags
  radix_select_kernel<<<BB, 1024, 0, stream>>>(score_u, LK, Tl, ttl);
  compact_threshold_kernel<<<BB, 1024, 0, stream>>>(score_u, Tl, ttl,
                                                    learned_list, o_lidx, LK, flag);
  // random top-4096 excluding learned
  rand_mask_kernel<<<RB, 256, 0, stream>>>(rnd, flag, rand_u);
  radix_select_kernel<<<BB, 1024, 0, stream>>>(rand_u, LK, Tr, ttr);
  compact_threshold_kernel<<<BB, 1024, 0, stream>>>(rand_u, Tr, ttr,
                                                    nullptr, nullptr, LK, flag);
  // combined = sorted union of both flag sets
  compact_flag_kernel<<<BB, 1024, 0, stream>>>(flag, comb_list, o_comb, K2);

  gather_kernel<<<(BB * K2 + 255) / 256, 256, 0, stream>>>(xyz, feat, comb_list,
                                                           o_feat, o_xyz);
}